// SPDBatchNormImpl_47742856463146
// MI455X (gfx1250) — compile-verified
//
#include <hip/hip_runtime.h>
#include <math.h>

// ---------------------------------------------------------------------------
// Hyperbolic (Lorentz) batch norm for MI455X / gfx1250.
// Persistent cooperative kernel; fp32 WMMA (v_wmma_f32_16x16x4_f32) for the
// per-iteration alpha dot-products and weighted-sum reductions.
// Round 2: 4-way independent WMMA accumulators for the alpha chain (ILP),
// float4/b128 vectorized final pass.
// ---------------------------------------------------------------------------

#define NPOINTS   100352            // 32*56*56
#define NCH       64
#define NBLOCKS   128
#define NTHREADS  256
#define NWAVES    (NTHREADS / 32)
#define PPB       (NPOINTS / NBLOCKS)   // 784 points per block (exact)
#define TILES_PB  (PPB / 16)            // 49 tiles of 16 points (exact)
#define KITERS    100
#define ETA_F     1.0f
#define BN_EPS_F  1e-5f
#define TINYF     1e-7f

static_assert(PPB * NBLOCKS == NPOINTS, "exact partition");
static_assert(TILES_PB * 16 == PPB, "exact tiles");

typedef __attribute__((ext_vector_type(2))) float v2f;
typedef __attribute__((ext_vector_type(4))) float v4f;
typedef __attribute__((ext_vector_type(8))) float v8f;

// ---- workspace layout (float offsets) ----
#define WS_MU     0      // [64]  current Karcher mean
#define WS_RM     64     // [64]  scratch mean m / final rm (geodesic result)
#define WS_FAC    130    // [1]   1/sqrt(rv + eps)
#define WS_CNT    160    // int: barrier arrival counter (monotonic)
#define WS_GEN    161    // int: barrier generation
#define WS_PART   192    // [NBLOCKS * PART_STRIDE] block partials (S[64], T)
#define PART_STRIDE 72

// ---------------------------------------------------------------------------
// Deterministic grid-wide barrier (monotonic counter, no reset race).
// ---------------------------------------------------------------------------
__device__ __forceinline__ void grid_sync(unsigned* cnt, unsigned* gen) {
  __threadfence();          // make this thread's global writes visible
  __syncthreads();
  if (threadIdx.x == 0) {
    unsigned g = __hip_atomic_load(gen, __ATOMIC_ACQUIRE, __HIP_MEMORY_SCOPE_AGENT);
    unsigned a = __hip_atomic_fetch_add(cnt, 1u, __ATOMIC_ACQ_REL, __HIP_MEMORY_SCOPE_AGENT) + 1u;
    if (a == (unsigned)NBLOCKS * (g + 1u)) {
      __hip_atomic_fetch_add(gen, 1u, __ATOMIC_ACQ_REL, __HIP_MEMORY_SCOPE_AGENT);
    } else {
      while (__hip_atomic_load(gen, __ATOMIC_ACQUIRE, __HIP_MEMORY_SCOPE_AGENT) == g) {
        __builtin_amdgcn_s_sleep(2);
      }
    }
  }
  __syncthreads();
}

// ---------------------------------------------------------------------------
// Exact-fp32 WMMA: D(16x16) = A(16x4) * B(4x16) + C.
// ---------------------------------------------------------------------------
__device__ __forceinline__ v8f wmma_f32(v2f a, v2f b, v8f c) {
  return __builtin_amdgcn_wmma_f32_16x16x4_f32(false, a, false, b, (short)0, c,
                                               false, false);
}

// alpha for a 16-point tile: result vgpr r holds alpha[point r] in lanes 0-15
// and alpha[point 8+r] in lanes 16-31 (all columns equal: B cols are equal).
// 4 independent accumulator chains (dependency distance 4) for XDL ILP.
__device__ __forceinline__ v8f alpha_tile(const float* __restrict__ Pt,
                                          const v2f (&Bnu)[16],
                                          int l15, int half) {
  const v8f z = {0.f, 0.f, 0.f, 0.f, 0.f, 0.f, 0.f, 0.f};
  v8f D0 = z, D1 = z, D2 = z, D3 = z;
  const float* rowp = Pt + l15 * NCH + 2 * half;  // A-layout: K0/K1 | K2/K3
#pragma unroll
  for (int k = 0; k < 4; ++k) {
    v2f A0 = *(const v2f*)(rowp + 4 * (4 * k + 0));
    v2f A1 = *(const v2f*)(rowp + 4 * (4 * k + 1));
    v2f A2 = *(const v2f*)(rowp + 4 * (4 * k + 2));
    v2f A3 = *(const v2f*)(rowp + 4 * (4 * k + 3));
    D0 = wmma_f32(A0, Bnu[4 * k + 0], D0);
    D1 = wmma_f32(A1, Bnu[4 * k + 1], D1);
    D2 = wmma_f32(A2, Bnu[4 * k + 2], D2);
    D3 = wmma_f32(A3, Bnu[4 * k + 3], D3);
  }
  return (D0 + D1) + (D2 + D3);
}

// ---------------------------------------------------------------------------
__global__ void hypbn_init(float* W) {
  if (threadIdx.x == 0) {
    ((unsigned*)(W + WS_CNT))[0] = 0u;
    ((unsigned*)(W + WS_GEN))[0] = 0u;
  }
}

__global__ __launch_bounds__(NTHREADS)
void hypbn_persistent(const float* __restrict__ X,
                      const float* __restrict__ rmean,
                      const float* __restrict__ rvar,
                      float* __restrict__ out,
                      float* W) {
  __shared__ __attribute__((aligned(16))) float s_nu[NCH];
  __shared__ __attribute__((aligned(16))) float s_red[NWAVES * 65];

  const int b    = blockIdx.x;
  const int t    = threadIdx.x;
  const int wv   = t >> 5;
  const int lane = t & 31;
  const int half = lane >> 4;
  const int l15  = lane & 15;

  unsigned* cnt = (unsigned*)(W + WS_CNT);
  unsigned* gen = (unsigned*)(W + WS_GEN);
  const float* Xb = X + (size_t)b * PPB * NCH;
  float* part = W + WS_PART + b * PART_STRIDE;

  // ================= Phase 0: Euclidean mean -> hyperboloid ================
  {
    const int ch = t & 63, grp = t >> 6;  // 4 point-groups x 64 channels
    float acc = 0.f;
    for (int p = grp; p < PPB; p += 4) acc += Xb[p * NCH + ch];
    s_red[t] = acc;
    __syncthreads();
    if (t < 64)
      part[t] = s_red[t] + s_red[t + 64] + s_red[t + 128] + s_red[t + 192];
    grid_sync(cnt, gen);
    if (b == 0) {
      if (t < 64) {
        float s = 0.f;
        for (int j = 0; j < NBLOCKS; ++j) s += W[WS_PART + j * PART_STRIDE + t];
        W[WS_RM + t] = s * (1.0f / (float)NPOINTS);   // m = mean(X)
      }
      __syncthreads();
      if (t == 0) {
        float ss = 0.f;
        for (int c = 0; c < NCH; ++c) { float m = W[WS_RM + c]; ss += m * m; }
        float m0 = W[WS_RM];
        float q = 2.f * m0 * m0 - ss;                 // -mink(m,m)
        float inv = 1.0f / sqrtf(fmaxf(q, TINYF));
        for (int c = 0; c < NCH; ++c) W[WS_MU + c] = W[WS_RM + c] * inv;
      }
    }
    grid_sync(cnt, gen);
  }

  // ================= Phase 1: Karcher (Frechet mean) iterations ============
  for (int it = 0; it < KITERS; ++it) {
    // nu = (mu0, -mu1, ..., -mu63): alpha_p = -mink(mu, X_p) = sum_c X_pc*nu_c
    if (t < NCH) { float m = W[WS_MU + t]; s_nu[t] = (t == 0) ? m : -m; }
    __syncthreads();

    // B-layout broadcast of nu: row k of B(4x16) = VGPR(k%2), halfwave(k/2)
    v2f Bnu[16];
#pragma unroll
    for (int k = 0; k < 16; ++k) {
      Bnu[k].x = s_nu[4 * k + 2 * half];
      Bnu[k].y = s_nu[4 * k + 2 * half + 1];
    }

    v8f accS[4];
#pragma unroll
    for (int g = 0; g < 4; ++g)
      accS[g] = (v8f){0.f, 0.f, 0.f, 0.f, 0.f, 0.f, 0.f, 0.f};
    float regT = 0.f;  // sum of w_p * alpha_p (per half-wave subset)

    for (int tile = wv; tile < TILES_PB; tile += NWAVES) {
      const float* Pt = Xb + (size_t)tile * 16 * NCH;
      if (tile + NWAVES < TILES_PB)
        __builtin_prefetch(Pt + (size_t)NWAVES * 16 * NCH, 0, 1);

      v8f Da = alpha_tile(Pt, Bnu, l15, half);

      // weights w = arccosh(a)/sqrt(a^2-1); lanes0-15: pts 0-7, lanes16-31: 8-15
      float wloc[8];
#pragma unroll
      for (int r = 0; r < 8; ++r) {
        float a   = fmaxf(Da[r], 1.0f + TINYF);
        float srt = sqrtf(a * a - 1.0f);
        float d   = logf(a + srt);          // arccosh
        float wgt = d / srt;
        wloc[r] = wgt;
        regT += wgt * a;
      }
      // broadcast all 16 point-weights to every lane
      float wp[16];
#pragma unroll
      for (int r = 0; r < 8; ++r) {
        wp[r]     = __shfl(wloc[r], 0, 32);
        wp[r + 8] = __shfl(wloc[r], 16, 32);
      }
      // S[c] += sum_p w_p * X[p][c] via WMMA: A = weights, B = X (4pt x 16ch)
      // g-inner order keeps the accS[0..3] chains at dependency distance 4.
#pragma unroll
      for (int q4 = 0; q4 < 4; ++q4) {
        v2f Aw;
        Aw.x = half ? wp[4 * q4 + 2] : wp[4 * q4 + 0];
        Aw.y = half ? wp[4 * q4 + 3] : wp[4 * q4 + 1];
#pragma unroll
        for (int g = 0; g < 4; ++g) {
          v2f Bx;
          Bx.x = Pt[(4 * q4 + 2 * half)     * NCH + g * 16 + l15];
          Bx.y = Pt[(4 * q4 + 2 * half + 1) * NCH + g * 16 + l15];
          accS[g] = wmma_f32(Aw, Bx, accS[g]);
        }
      }
    }

    // wave partials -> LDS -> block partial -> global slot
    float Tw = __shfl(regT, 0, 32) + __shfl(regT, 16, 32);
    if (lane < 16) {
#pragma unroll
      for (int g = 0; g < 4; ++g) s_red[wv * 65 + g * 16 + l15] = accS[g][0];
    }
    if (lane == 0) s_red[wv * 65 + 64] = Tw;
    __syncthreads();
    if (t < 65) {
      float s = 0.f;
#pragma unroll
      for (int w = 0; w < NWAVES; ++w) s += s_red[w * 65 + t];
      part[t] = s;
    }
    grid_sync(cnt, gen);

    // block 0: deterministic combine + expmap update of mu
    if (b == 0) {
      if (t < 65) {
        float s = 0.f;
        for (int j = 0; j < NBLOCKS; ++j) s += W[WS_PART + j * PART_STRIDE + t];
        s_red[t] = s;
      }
      __syncthreads();
      if (t == 0) {
        const float invN = 1.0f / (float)NPOINTS;
        float Tm = s_red[64] * invN;
        float q = 0.f;
        for (int c = 0; c < NCH; ++c) {
          float gc = s_red[c] * invN - Tm * W[WS_MU + c];  // g = mean(logmap)
          s_red[128 + c] = gc;
          q += gc * gc;
        }
        q -= 2.f * s_red[128] * s_red[128];                // mink(g,g)
        float nv = sqrtf(fmaxf(q, 0.f));
        float ch = coshf(nv);
        float sh = (nv > TINYF) ? (sinhf(nv) / fmaxf(nv, TINYF)) : 1.0f;
        for (int c = 0; c < NCH; ++c)
          W[WS_MU + c] = ch * W[WS_MU + c] + sh * s_red[128 + c];  // expmap
      }
    }
    grid_sync(cnt, gen);
  }

  // ====== Phase 2: Frechet variance (vs mu) + geodesic(running_mean, mu) ===
  {
    if (t < NCH) { float m = W[WS_MU + t]; s_nu[t] = (t == 0) ? m : -m; }
    __syncthreads();
    v2f Bnu[16];
#pragma unroll
    for (int k = 0; k < 16; ++k) {
      Bnu[k].x = s_nu[4 * k + 2 * half];
      Bnu[k].y = s_nu[4 * k + 2 * half + 1];
    }
    float vac = 0.f;
    for (int tile = wv; tile < TILES_PB; tile += NWAVES) {
      const float* Pt = Xb + (size_t)tile * 16 * NCH;
      v8f Da = alpha_tile(Pt, Bnu, l15, half);
#pragma unroll
      for (int r = 0; r < 8; ++r) {
        float a = fmaxf(Da[r], 1.0f + TINYF);
        float d = logf(a + sqrtf(a * a - 1.0f));
        vac += d * d;
      }
    }
    float Tw = __shfl(vac, 0, 32) + __shfl(vac, 16, 32);
    if (lane == 0) s_red[wv] = Tw;
    __syncthreads();
    if (t == 0) {
      float s = 0.f;
      for (int w = 0; w < NWAVES; ++w) s += s_red[w];
      part[0] = s;
    }
    grid_sync(cnt, gen);
    if (b == 0 && t == 0) {
      float s = 0.f;
      for (int j = 0; j < NBLOCKS; ++j) s += W[WS_PART + j * PART_STRIDE];
      float var = s / (float)NPOINTS;
      float rv = (1.0f - ETA_F) * rvar[0] + ETA_F * var;
      W[WS_FAC] = 1.0f / sqrtf(rv + BN_EPS_F);

      // rm = expmap(rmean, ETA * logmap(rmean, mu))
      float dotv = 0.f;
      for (int c = 0; c < NCH; ++c) dotv += rmean[c] * W[WS_MU + c];
      float alpha = fmaxf(2.f * rmean[0] * W[WS_MU] - dotv, 1.0f + TINYF);
      float srt = sqrtf(alpha * alpha - 1.0f);
      float d   = logf(alpha + srt);
      float fac = ETA_F * d / srt;
      float q = 0.f;
      for (int c = 0; c < NCH; ++c) {
        float gc = fac * (W[WS_MU + c] - alpha * rmean[c]);
        s_red[c] = gc;
        q += gc * gc;
      }
      q -= 2.f * s_red[0] * s_red[0];
      float nv = sqrtf(fmaxf(q, 0.f));
      float chh = coshf(nv);
      float shh = (nv > TINYF) ? (sinhf(nv) / fmaxf(nv, TINYF)) : 1.0f;
      for (int c = 0; c < NCH; ++c)
        W[WS_RM + c] = chh * rmean[c] + shh * s_red[c];
    }
    grid_sync(cnt, gen);
  }

  // ====== Phase 3: gyrotrans(gyroinv(rm), X) + gyroscalarprod(., factor) ===
  {
    if (t < NCH) s_nu[t] = (t == 0) ? 0.f : -W[WS_RM + t];  // as_c, as_0 = 0
    __syncthreads();
    const float a0 = W[WS_RM];
    const float r  = W[WS_FAC];
    const float inv1pa0 = 1.0f / (1.0f + a0);
    float* Ob = out + (size_t)b * PPB * NCH;
    for (int p = t; p < PPB; p += NTHREADS) {
      const float* xp = Xb + (size_t)p * NCH;
      float* op = Ob + (size_t)p * NCH;
      // dot = sum_{c>=1} X[c] * as[c]   (s_nu[0] == 0)
      float dot = 0.f;
#pragma unroll
      for (int c = 0; c < NCH; c += 4) {
        v4f xv = *(const v4f*)(xp + c);
        v4f av = *(const v4f*)(s_nu + c);
        dot += xv.x * av.x + xv.y * av.y + xv.z * av.z + xv.w * av.w;
      }
      float x0 = xp[0];
      float t0 = a0 * x0 + dot;
      float u  = x0 + dot * inv1pa0;
      float y0 = fmaxf(t0, 1.0f + TINYF);
      float den = sqrtf(y0 * y0 - 1.0f);
      float dd  = logf(y0 + den);
      float rd  = r * dd;
      float fc  = (den > 1e-6f) ? (sinhf(rd) / fmaxf(den, 1e-6f)) : r;
      float o0  = coshf(rd);
#pragma unroll
      for (int c = 0; c < NCH; c += 4) {
        v4f xv = *(const v4f*)(xp + c);
        v4f av = *(const v4f*)(s_nu + c);
        v4f ov = (xv + av * u) * fc;
        if (c == 0) ov.x = o0;
        *(v4f*)(op + c) = ov;
      }
    }
  }
}

// ---------------------------------------------------------------------------
extern "C" void kernel_launch(void* const* d_in, const int* in_sizes, int n_in,
                              void* d_out, int out_size, void* d_ws, size_t ws_size,
                              hipStream_t stream) {
  (void)in_sizes; (void)n_in; (void)out_size; (void)ws_size;
  const float* X     = (const float*)d_in[0];
  const float* rmean = (const float*)d_in[1];
  const float* rvar  = (const float*)d_in[2];
  float* out = (float*)d_out;
  float* W   = (float*)d_ws;

  hypbn_init<<<1, 64, 0, stream>>>(W);
  hypbn_persistent<<<NBLOCKS, NTHREADS, 0, stream>>>(X, rmean, rvar, out, W);
}